// DGCNN_29128468201628
// MI455X (gfx1250) — compile-verified
//
#include <hip/hip_runtime.h>
#include <math.h>

typedef __attribute__((ext_vector_type(2))) float v2f;
typedef __attribute__((ext_vector_type(8))) float v8f;

#define BATCH 8
#define NPTS  1024
#define KNN   20
#define NEG_HUGE (-3.402823466e38f)
#define EPSBN 1e-5f

// ---------------------------------------------------------------------------
// ordered-int float atomic max (works for mixed signs when init <= all values)
// ---------------------------------------------------------------------------
__device__ __forceinline__ void atomicMaxF(float* addr, float v) {
    if (v >= 0.0f) atomicMax((int*)addr, __float_as_int(v));
    else           atomicMin((unsigned int*)addr, __float_as_uint(v));
}

// ---------------------------------------------------------------------------
// small utility kernels
// ---------------------------------------------------------------------------
__global__ void k_fill(float* p, float v, int n) {
    int i = blockIdx.x * blockDim.x + threadIdx.x;
    if (i < n) p[i] = v;
}

// fill strided channel region: base already points at column offset
__global__ void k_fill_strided(float* base, int rows, int rowStride, int ncol, float v) {
    int i = blockIdx.x * blockDim.x + threadIdx.x;
    if (i < rows * ncol) {
        int r = i / ncol, c = i - r * ncol;
        base[r * rowStride + c] = v;
    }
}

// pad W[O][K2] -> Wp[O][Kp] (zero fill K2..Kp)
__global__ void k_pad_w(const float* __restrict__ W, float* __restrict__ Wp,
                        int O, int K2, int Kp) {
    int i = blockIdx.x * blockDim.x + threadIdx.x;
    if (i < O * Kp) {
        int o = i / Kp, c = i - o * Kp;
        Wp[i] = (c < K2) ? W[o * K2 + c] : 0.f;
    }
}

// x [B,3,N] -> xt [B,N,3]
__global__ void k_transpose(const float* __restrict__ x, float* __restrict__ xt) {
    int i = blockIdx.x * blockDim.x + threadIdx.x;
    if (i < BATCH * 3 * NPTS) {
        int b = i / (3 * NPTS);
        int rem = i - b * 3 * NPTS;
        int c = rem / NPTS, n = rem - c * NPTS;
        xt[(b * NPTS + n) * 3 + c] = x[i];
    }
}

// ---------------------------------------------------------------------------
// kNN: one wave (32 lanes) per point. neg_dist = 2*inner - |xm|^2 - |xn|^2
// ---------------------------------------------------------------------------
__global__ __launch_bounds__(32) void k_knn(const float* __restrict__ F, int Fstride,
                                            int Cin, int* __restrict__ idx) {
    int r = blockIdx.x;                 // global row = b*N + n
    int lane = threadIdx.x;
    __shared__ float center[128];
    __shared__ float mv[32 * KNN];
    __shared__ int   mi[32 * KNN];

    for (int c = lane; c < Cin; c += 32) center[c] = F[r * Fstride + c];
    __syncthreads();

    float xxn = 0.f;
    for (int c = 0; c < Cin; ++c) xxn += center[c] * center[c];

    int rowBase = (r >> 10) << 10;      // b * N

    float bv[KNN]; int bi[KNN];
#pragma unroll
    for (int s = 0; s < KNN; ++s) { bv[s] = NEG_HUGE; bi[s] = 0; }

    for (int j = 0; j < NPTS / 32; ++j) {
        int m = lane + (j << 5);
        const float* Fm = F + (size_t)(rowBase + m) * Fstride;
        float inner = 0.f, xm = 0.f;
        for (int c = 0; c < Cin; ++c) { float t = Fm[c]; inner += center[c] * t; xm += t * t; }
        float v = 2.f * inner - xm - xxn;
        int id = m;
#pragma unroll
        for (int s = 0; s < KNN; ++s) {
            if (v > bv[s]) {
                float tv = bv[s]; bv[s] = v; v = tv;
                int ti = bi[s]; bi[s] = id; id = ti;
            }
        }
    }
#pragma unroll
    for (int s = 0; s < KNN; ++s) { mv[lane * KNN + s] = bv[s]; mi[lane * KNN + s] = bi[s]; }
    __syncthreads();

    if (lane == 0) {
        float tv[KNN]; int ti[KNN];
#pragma unroll
        for (int s = 0; s < KNN; ++s) { tv[s] = NEG_HUGE; ti[s] = 0; }
        for (int e = 0; e < 32 * KNN; ++e) {
            float v = mv[e]; int id = mi[e];
#pragma unroll
            for (int s = 0; s < KNN; ++s) {
                if (v > tv[s]) {
                    float a = tv[s]; tv[s] = v; v = a;
                    int c = ti[s]; ti[s] = id; id = c;
                }
            }
        }
#pragma unroll
        for (int s = 0; s < KNN; ++s) idx[r * KNN + s] = ti[s];
    }
}

// ---------------------------------------------------------------------------
// WMMA f32 16x16x4 tile kernel core: unconditional 8B vector loads.
// A fragment: VGPR0/1 = K {k0, k0+1} (lanes 0-15) / {k0+2, k0+3} (lanes 16-31),
// M = lane&15.  B fragment: same K pattern, N-col = lane&15.
// Kp must be a multiple of 4; A tile in LDS and Wp rows are zero-padded.
// ---------------------------------------------------------------------------
__device__ __forceinline__ v8f wmma_tile_f32(const float* At, const float* Wp,
                                             int Kp, int o0, int lane) {
    int col = lane & 15;
    int koff = (lane >> 4) << 1;        // 0 or 2
    const float* Arow = At + col * Kp + koff;
    const float* Wrow = Wp + (size_t)(o0 + col) * Kp + koff;
    v8f acc = {0.f, 0.f, 0.f, 0.f, 0.f, 0.f, 0.f, 0.f};
#pragma unroll 4
    for (int k0 = 0; k0 < Kp; k0 += 4) {
        v2f a = *reinterpret_cast<const v2f*>(Arow + k0);
        v2f b = *reinterpret_cast<const v2f*>(Wrow + k0);
        acc = __builtin_amdgcn_wmma_f32_16x16x4_f32(false, a, false, b,
                                                    (short)0, acc, false, false);
    }
    return acc;
}

// ---------------------------------------------------------------------------
// Edge-conv GEMM. Block = 16 edge rows staged in LDS, 4 waves split O tiles.
// Pass 1: accumulate per-channel sum / sumsq of h = edge @ W^T.
// Pass 2: recompute, apply BN affine + LeakyReLU, atomic-max over k into feats.
// ---------------------------------------------------------------------------
__device__ __forceinline__ void edge_build_tile(float* At, int* rN, int* rNbr,
                                                const float* F, int Fstride, int Cin,
                                                const int* idx, int r0, int tid, int Kp) {
    if (tid < 16) {
        int r = r0 + tid;
        int bn = r / KNN;
        int kk = r - bn * KNN;
        rN[tid] = bn;
        rNbr[tid] = ((bn >> 10) << 10) + idx[bn * KNN + kk];
    }
    __syncthreads();
    int K2 = 2 * Cin;
    int total = 16 * Kp;
    for (int e = tid; e < total; e += 128) {
        int j = e / Kp, c = e - j * Kp;
        float v = 0.f;
        if (c < Cin)      v = F[(size_t)rNbr[j] * Fstride + c] - F[(size_t)rN[j] * Fstride + c];
        else if (c < K2)  v = F[(size_t)rN[j] * Fstride + (c - Cin)];
        At[j * Kp + c] = v;
    }
    __syncthreads();
}

__global__ __launch_bounds__(128) void k_edge_stats(const float* __restrict__ F, int Fstride,
                                                    int Cin, const int* __restrict__ idx,
                                                    const float* __restrict__ Wp, int Kp,
                                                    int O,
                                                    float* __restrict__ sum,
                                                    float* __restrict__ sumsq) {
    __shared__ float At[16 * 512];
    __shared__ int rN[16], rNbr[16];
    int tid = threadIdx.x;
    edge_build_tile(At, rN, rNbr, F, Fstride, Cin, idx, blockIdx.x * 16, tid, Kp);

    int wave = tid >> 5, lane = tid & 31;
    int col = lane & 15, hs = lane >> 4;
    int numOT = O >> 4;
    for (int ot = wave; ot < numOT; ot += 4) {
        int o0 = ot << 4;
        v8f acc = wmma_tile_f32(At, Wp, Kp, o0, lane);
        float s = 0.f, sq = 0.f;
#pragma unroll
        for (int j = 0; j < 8; ++j) { float h = acc[j]; s += h; sq += h * h; }
        s  += __shfl_xor(s, 16);
        sq += __shfl_xor(sq, 16);
        if (hs == 0) {
            atomicAdd(&sum[o0 + col], s);
            atomicAdd(&sumsq[o0 + col], sq);
        }
    }
}

__global__ __launch_bounds__(128) void k_edge_apply(const float* __restrict__ F, int Fstride,
                                                    int Cin, const int* __restrict__ idx,
                                                    const float* __restrict__ Wp, int Kp,
                                                    int O,
                                                    const float* __restrict__ scale,
                                                    const float* __restrict__ shift,
                                                    float* __restrict__ outF, int outStride,
                                                    int oOff) {
    __shared__ float At[16 * 512];
    __shared__ int rN[16], rNbr[16];
    int tid = threadIdx.x;
    edge_build_tile(At, rN, rNbr, F, Fstride, Cin, idx, blockIdx.x * 16, tid, Kp);

    int wave = tid >> 5, lane = tid & 31;
    int col = lane & 15, hs = lane >> 4;
    int numOT = O >> 4;
    for (int ot = wave; ot < numOT; ot += 4) {
        int o0 = ot << 4;
        v8f acc = wmma_tile_f32(At, Wp, Kp, o0, lane);
        int oc = o0 + col;
        float sc = scale[oc], sh = shift[oc];
#pragma unroll
        for (int j = 0; j < 8; ++j) {
            float h = acc[j] * sc + sh;
            h = (h >= 0.f) ? h : 0.2f * h;
            int tr = j + (hs << 3);
            atomicMaxF(&outF[(size_t)rN[tr] * outStride + oOff + oc], h);
        }
    }
}

// ---------------------------------------------------------------------------
// Global 512 -> 1024 conv (plain rows, K exactly 512 so Wg used unpadded);
// apply pass fuses max-pool (atomic max) and avg-pool (atomic add) over N.
// ---------------------------------------------------------------------------
__global__ __launch_bounds__(128) void k_glob_stats(const float* __restrict__ F,
                                                    const float* __restrict__ W,
                                                    float* __restrict__ sum,
                                                    float* __restrict__ sumsq) {
    const int Kp = 512, O = 1024;
    __shared__ float At[16 * 512];
    int tid = threadIdx.x;
    int r0 = blockIdx.x * 16;
    for (int e = tid; e < 16 * 512; e += 128) {
        int j = e >> 9, c = e & 511;
        At[e] = F[(size_t)(r0 + j) * 512 + c];
    }
    __syncthreads();

    int wave = tid >> 5, lane = tid & 31;
    int col = lane & 15, hs = lane >> 4;
    for (int ot = wave; ot < (O >> 4); ot += 4) {
        int o0 = ot << 4;
        v8f acc = wmma_tile_f32(At, W, Kp, o0, lane);
        float s = 0.f, sq = 0.f;
#pragma unroll
        for (int j = 0; j < 8; ++j) { float h = acc[j]; s += h; sq += h * h; }
        s  += __shfl_xor(s, 16);
        sq += __shfl_xor(sq, 16);
        if (hs == 0) {
            atomicAdd(&sum[o0 + col], s);
            atomicAdd(&sumsq[o0 + col], sq);
        }
    }
}

__global__ __launch_bounds__(128) void k_glob_apply(const float* __restrict__ F,
                                                    const float* __restrict__ W,
                                                    const float* __restrict__ scale,
                                                    const float* __restrict__ shift,
                                                    float* __restrict__ gmax,
                                                    float* __restrict__ gsum) {
    const int Kp = 512, O = 1024;
    __shared__ float At[16 * 512];
    int tid = threadIdx.x;
    int r0 = blockIdx.x * 16;
    for (int e = tid; e < 16 * 512; e += 128) {
        int j = e >> 9, c = e & 511;
        At[e] = F[(size_t)(r0 + j) * 512 + c];
    }
    __syncthreads();

    int wave = tid >> 5, lane = tid & 31;
    int col = lane & 15, hs = lane >> 4;
    for (int ot = wave; ot < (O >> 4); ot += 4) {
        int o0 = ot << 4;
        v8f acc = wmma_tile_f32(At, W, Kp, o0, lane);
        int oc = o0 + col;
        float sc = scale[oc], sh = shift[oc];
#pragma unroll
        for (int j = 0; j < 8; ++j) {
            float h = acc[j] * sc + sh;
            h = (h >= 0.f) ? h : 0.2f * h;
            int r = r0 + j + (hs << 3);   // b*N + n
            int b = r >> 10;
            atomicMaxF(&gmax[b * 1024 + oc], h);
            atomicAdd(&gsum[b * 1024 + oc], h);
        }
    }
}

// ---------------------------------------------------------------------------
// BN finalize: scale = gamma * rsqrt(var + eps), shift = beta - mean * scale
// ---------------------------------------------------------------------------
__global__ void k_bn_finalize(const float* sum, const float* sumsq,
                              const float* gamma, const float* beta,
                              float* scale, float* shift, int O, float invCnt) {
    int o = blockIdx.x * blockDim.x + threadIdx.x;
    if (o < O) {
        float m = sum[o] * invCnt;
        float var = sumsq[o] * invCnt - m * m;
        float sc = gamma[o] * rsqrtf(var + EPSBN);
        scale[o] = sc;
        shift[o] = beta[o] - m * sc;
    }
}

// z[b,0:1024]=gmax, z[b,1024:2048]=gsum/N
__global__ void k_buildz(const float* gmax, const float* gsum, float* z) {
    int i = blockIdx.x * blockDim.x + threadIdx.x;
    if (i < BATCH * 1024) {
        int b = i >> 10, o = i & 1023;
        z[b * 2048 + o]        = gmax[i];
        z[b * 2048 + 1024 + o] = gsum[i] * (1.f / (float)NPTS);
    }
}

// ---------------------------------------------------------------------------
// FC layers (M = 8 rows): one thread per output channel, BN over batch.
// ---------------------------------------------------------------------------
__global__ void k_fc_bn(const float* __restrict__ Z, int In,
                        const float* __restrict__ W, const float* __restrict__ bias,
                        const float* __restrict__ gamma, const float* __restrict__ beta,
                        float* __restrict__ out, int O) {
    int o = blockIdx.x * blockDim.x + threadIdx.x;
    if (o >= O) return;
    float h[BATCH];
    const float* w = W + (size_t)o * In;
    for (int b = 0; b < BATCH; ++b) {
        float s = bias ? bias[o] : 0.f;
        const float* zr = Z + (size_t)b * In;
        for (int i = 0; i < In; ++i) s += zr[i] * w[i];
        h[b] = s;
    }
    float m = 0.f;
#pragma unroll
    for (int b = 0; b < BATCH; ++b) m += h[b];
    m *= (1.f / BATCH);
    float var = 0.f;
#pragma unroll
    for (int b = 0; b < BATCH; ++b) { float d = h[b] - m; var += d * d; }
    var *= (1.f / BATCH);
    float sc = gamma[o] * rsqrtf(var + EPSBN);
    float sh = beta[o] - m * sc;
#pragma unroll
    for (int b = 0; b < BATCH; ++b) {
        float v = h[b] * sc + sh;
        out[b * O + o] = (v >= 0.f) ? v : 0.2f * v;
    }
}

__global__ void k_fc_final(const float* __restrict__ Z, int In,
                           const float* __restrict__ W, const float* __restrict__ bias,
                           float* __restrict__ out, int O) {
    int o = blockIdx.x * blockDim.x + threadIdx.x;
    if (o >= O) return;
    const float* w = W + (size_t)o * In;
    for (int b = 0; b < BATCH; ++b) {
        float s = bias[o];
        const float* zr = Z + (size_t)b * In;
        for (int i = 0; i < In; ++i) s += zr[i] * w[i];
        out[b * O + o] = s;
    }
}

// ---------------------------------------------------------------------------
// host-side driver
// ---------------------------------------------------------------------------
static void run_edge_layer(const float* F, int Fstride, int Cin, int O,
                           const float* W, const float* g, const float* bt,
                           int* idxb, float* feats, int oOff, float* Wp,
                           float* sum, float* sumsq, float* scale, float* shift,
                           hipStream_t s) {
    const int M = BATCH * NPTS * KNN;             // 163840, multiple of 16
    const int K2 = 2 * Cin;
    const int Kp = (K2 + 3) & ~3;
    k_knn<<<BATCH * NPTS, 32, 0, s>>>(F, Fstride, Cin, idxb);
    k_pad_w<<<(O * Kp + 255) / 256, 256, 0, s>>>(W, Wp, O, K2, Kp);
    k_fill<<<(2048 + 255) / 256, 256, 0, s>>>(sum, 0.f, 2048);   // sum+sumsq contiguous
    k_edge_stats<<<M / 16, 128, 0, s>>>(F, Fstride, Cin, idxb, Wp, Kp, O, sum, sumsq);
    k_bn_finalize<<<(O + 63) / 64, 64, 0, s>>>(sum, sumsq, g, bt, scale, shift, O,
                                               1.f / (float)M);
    k_fill_strided<<<(BATCH * NPTS * O + 255) / 256, 256, 0, s>>>(feats + oOff,
                                                                  BATCH * NPTS, 512, O,
                                                                  NEG_HUGE);
    k_edge_apply<<<M / 16, 128, 0, s>>>(F, Fstride, Cin, idxb, Wp, Kp, O, scale, shift,
                                        feats, 512, oOff);
}

extern "C" void kernel_launch(void* const* d_in, const int* in_sizes, int n_in,
                              void* d_out, int out_size, void* d_ws, size_t ws_size,
                              hipStream_t stream) {
    (void)in_sizes; (void)n_in; (void)out_size; (void)ws_size;
    const float* x   = (const float*)d_in[0];
    const float* W0  = (const float*)d_in[1];
    const float* g0  = (const float*)d_in[2];
    const float* b0  = (const float*)d_in[3];
    const float* W1  = (const float*)d_in[4];
    const float* g1  = (const float*)d_in[5];
    const float* b1  = (const float*)d_in[6];
    const float* W2  = (const float*)d_in[7];
    const float* g2  = (const float*)d_in[8];
    const float* b2  = (const float*)d_in[9];
    const float* W3  = (const float*)d_in[10];
    const float* g3  = (const float*)d_in[11];
    const float* b3  = (const float*)d_in[12];
    const float* Wg  = (const float*)d_in[13];
    const float* gg  = (const float*)d_in[14];
    const float* bg  = (const float*)d_in[15];
    const float* Wf1 = (const float*)d_in[16];
    const float* gf1 = (const float*)d_in[17];
    const float* bf1 = (const float*)d_in[18];
    const float* Wf2 = (const float*)d_in[19];
    const float* bf2w= (const float*)d_in[20];
    const float* gf2 = (const float*)d_in[21];
    const float* bf2 = (const float*)d_in[22];
    const float* Wf3 = (const float*)d_in[23];
    const float* bf3w= (const float*)d_in[24];
    float* out = (float*)d_out;

    // workspace carve-up (floats)
    float* ws    = (float*)d_ws;
    float* xt    = ws;                     // 24576
    float* feats = ws + 24576;             // 4194304  (B*N*512)
    float* sum   = feats + 4194304;        // 1024
    float* sumsq = sum + 1024;             // 1024  (contiguous with sum)
    float* scale = sumsq + 1024;           // 1024
    float* shift = scale + 1024;           // 1024
    float* gmax  = shift + 1024;           // 8192
    float* gsum  = gmax + 8192;            // 8192
    float* z     = gsum + 8192;            // 16384
    float* z1    = z + 16384;              // 4096
    float* z2    = z1 + 4096;              // 2048
    int*   idxb  = (int*)(z2 + 2048);      // 163840 ints
    float* Wp    = (float*)(idxb + 163840);// up to 256*256 floats (edge layers)

    k_transpose<<<(BATCH * 3 * NPTS + 255) / 256, 256, 0, stream>>>(x, xt);

    // edge conv stack; outputs written into concatenated feats channels
    run_edge_layer(xt,          3,   3,   64, W0, g0, b0, idxb, feats,   0, Wp, sum, sumsq, scale, shift, stream);
    run_edge_layer(feats + 0,   512, 64,  64, W1, g1, b1, idxb, feats,  64, Wp, sum, sumsq, scale, shift, stream);
    run_edge_layer(feats + 64,  512, 64, 128, W2, g2, b2, idxb, feats, 128, Wp, sum, sumsq, scale, shift, stream);
    run_edge_layer(feats + 128, 512, 128,256, W3, g3, b3, idxb, feats, 256, Wp, sum, sumsq, scale, shift, stream);

    // global conv 512 -> 1024 with fused pooling (K = 512 exactly: Wg unpadded)
    k_fill<<<(2048 + 255) / 256, 256, 0, stream>>>(sum, 0.f, 2048);
    k_glob_stats<<<BATCH * NPTS / 16, 128, 0, stream>>>(feats, Wg, sum, sumsq);
    k_bn_finalize<<<(1024 + 63) / 64, 64, 0, stream>>>(sum, sumsq, gg, bg, scale, shift,
                                                       1024, 1.f / (float)(BATCH * NPTS));
    k_fill<<<(8192 + 255) / 256, 256, 0, stream>>>(gmax, NEG_HUGE, 8192);
    k_fill<<<(8192 + 255) / 256, 256, 0, stream>>>(gsum, 0.f, 8192);
    k_glob_apply<<<BATCH * NPTS / 16, 128, 0, stream>>>(feats, Wg, scale, shift, gmax, gsum);

    k_buildz<<<(BATCH * 1024 + 255) / 256, 256, 0, stream>>>(gmax, gsum, z);

    // FC head
    k_fc_bn<<<(512 + 63) / 64, 64, 0, stream>>>(z,  2048, Wf1, nullptr, gf1, bf1, z1, 512);
    k_fc_bn<<<(256 + 63) / 64, 64, 0, stream>>>(z1, 512,  Wf2, bf2w,   gf2, bf2, z2, 256);
    k_fc_final<<<(67 + 63) / 64, 64, 0, stream>>>(z2, 256, Wf3, bf3w, out, 67);
}